// SleepRegulationModel_88321707475483
// MI455X (gfx1250) — compile-verified
//
#include <hip/hip_runtime.h>
#include <math.h>

// ---------------------------------------------------------------------------
// gfx1250 native tanh (TRANS-class v_tanh_f32). Fall back to ocml tanhf if the
// toolchain lacks the builtin.
// ---------------------------------------------------------------------------
#if defined(__has_builtin)
#  if __has_builtin(__builtin_amdgcn_tanhf)
#    define SRM_TANH(x) __builtin_amdgcn_tanhf(x)
#  endif
#endif
#ifndef SRM_TANH
#  define SRM_TANH(x) tanhf(x)
#endif

namespace srm {
// Fixed model constants (from setParams in the reference)
constexpr int   T = 10000;
constexpr int   B = 512;

constexpr float INV_TAU_W  = 1.0f / 1.5e6f;
constexpr float INV_TAU_N  = 1.0f / 6.0e5f;
constexpr float INV_TAU_R  = 1.0f / 6.0e4f;
constexpr float INV_TAU_E  = 1.0f / 2.5e4f;
constexpr float INV_TAU_G  = 1.0f / 1.0e4f;
constexpr float INV_TAU_A  = 1.0f / 1.0e4f;
constexpr float INV_TAU_HW = 1.0f / 3.483e7f;
constexpr float INV_TAU_HS = 1.0f / 3.06e7f;

constexpr float F_W_MAX_H = 6.5f * 0.5f;   // F_MAX * 0.5 pre-folded
constexpr float F_N_MAX_H = 5.0f * 0.5f;
constexpr float F_R_MAX_H = 5.0f * 0.5f;

constexpr float INV_ALPHA_W = 1.0f / 0.5f;
constexpr float INV_ALPHA_N = 1.0f / 0.175f;
constexpr float INV_ALPHA_R = 1.0f / 0.13f;

constexpr float BETA_W = -0.4f;
constexpr float BETA_R = -0.9f;

constexpr float INV_GAMMA_E = 1.0f / 5.0f;
constexpr float INV_GAMMA_G = 1.0f / 4.0f;
constexpr float INV_GAMMA_A = 1.0f / 2.0f;

constexpr float H_MAX   = 1.0f;
constexpr float THETA_W = 2.0f;
constexpr float KAPPA   = 1.5f;
} // namespace srm

__device__ __forceinline__ float srm_tanh(float x) { return SRM_TANH(x); }

// comb(v) = (-3 v0 + 2 v1 + 4 v2 + 2 v3 + v4) / 6
__device__ __forceinline__ float rk_comb(const float v[5]) {
  return (-3.0f * v[0] + 2.0f * v[1] + 4.0f * v[2] + 2.0f * v[3] + v[4]) *
         (1.0f / 6.0f);
}

__global__ __launch_bounds__(32, 1)
void SleepRegulationModel_88321707475483_kernel(
    const float* __restrict__ noise,   // [T, B]
    const float* __restrict__ p_gRRe, const float* __restrict__ p_gRWe,
    const float* __restrict__ p_gWNi, const float* __restrict__ p_gWRi,
    const float* __restrict__ p_gNRi, const float* __restrict__ p_gNWi,
    float* __restrict__ out)           // [T, 7, B]
{
  using namespace srm;
  const int b = blockIdx.x * blockDim.x + threadIdx.x;
  if (b >= B) return;

  // Coupling scalars (uniform across lanes; single load each)
  const float gRRe = p_gRRe[0];
  const float gRWe = p_gRWe[0];
  const float gWNi = p_gWNi[0];
  const float gWRi = p_gWRi[0];
  const float gNRi = p_gNRi[0];
  const float gNWi = p_gNWi[0];

  // Persistent state, lives in VGPRs for all 10000 steps
  float sfW = 6.0f, sfN = 1e-3f, sfR = 1e-3f;
  float scW = 0.9f, scN = 1e-3f, scR = 1e-3f;
  float sh  = 0.5f;

  // Software-pipelined noise stream: nu holds noise[t], nu_next is in flight.
  float nu = noise[b];

  for (int t = 0; t < T; ++t) {
    // Issue next iteration's load early; its latency hides under the RK math.
    const int tn = (t + 1 < T) ? (t + 1) : (T - 1);
    const float nu_next = noise[(size_t)tn * B + b];
    // Deep prefetch (emits gfx1250 global_prefetch_b8; speculative, no counter)
    const int tp = (t + 16 < T) ? (t + 16) : (T - 1);
    __builtin_prefetch(&noise[(size_t)tp * B + b], 0, 1);

    // RK stage arrays; index 0 = step-initial value.
    float fW[5], fN[5], fR[5], cW[5], cN[5], cR[5], hh[5];
    fW[0] = sfW; fN[0] = sfN; fR[0] = sfR;
    cW[0] = scW; cN[0] = scN; cR[0] = scR; hh[0] = sh;

    const float A_RK[4] = {0.5f, 0.5f, 1.0f, 1.0f};

#pragma unroll
    for (int n = 0; n < 4; ++n) {
      const float a = A_RK[n];

      const float I_W = gNWi * cN[n] + gRWe * cR[n] + nu;
      const float I_N = gWNi * cW[n] + nu;
      const float I_R = gWRi * cW[n] + gNRi * cN[n] + gRRe * cR[n] + nu;

      const float fWn =
          fW[0] + a * INV_TAU_W *
          (F_W_MAX_H * (1.0f + srm_tanh((I_W - BETA_W) * INV_ALPHA_W)) - fW[n]);
      const float fNn =
          fN[0] + a * INV_TAU_N *
          (F_N_MAX_H * (1.0f + srm_tanh((I_N + KAPPA * hh[n]) * INV_ALPHA_N)) - fN[n]);
      const float fRn =
          fR[0] + a * INV_TAU_R *
          (F_R_MAX_H * (1.0f + srm_tanh((I_R - BETA_R) * INV_ALPHA_R)) - fR[n]);

      // Note: cX updates use the *new* fX (index n+1), matching the reference.
      const float cWn = cW[0] + a * INV_TAU_E * (srm_tanh(fWn * INV_GAMMA_E) - cW[n]);
      const float cNn = cN[0] + a * INV_TAU_G * (srm_tanh(fNn * INV_GAMMA_G) - cN[n]);
      const float cRn = cR[0] + a * INV_TAU_A * (srm_tanh(fRn * INV_GAMMA_A) - cR[n]);

      // Smooth Heaviside pair shares one tanh: sigmoid(5x)=0.5(1+tanh(2.5x)),
      // and sigmoid(-5x) = 1 - sigmoid(5x).
      const float Hw = 0.5f * (1.0f + srm_tanh(2.5f * (fW[n] - THETA_W)));
      const float Hs = 1.0f - Hw;
      const float hn =
          hh[0] + a * ((H_MAX - hh[n]) * INV_TAU_HW * Hw - hh[n] * INV_TAU_HS * Hs);

      fW[n + 1] = fWn; fN[n + 1] = fNn; fR[n + 1] = fRn;
      cW[n + 1] = cWn; cN[n + 1] = cNn; cR[n + 1] = cRn;
      hh[n + 1] = hn;
    }

    sfW = rk_comb(fW); sfN = rk_comb(fN); sfR = rk_comb(fR);
    scW = rk_comb(cW); scN = rk_comb(cN); scR = rk_comb(cR);
    sh  = rk_comb(hh);

    // traj[t, s, b]; streamed once, no reuse -> non-temporal stores
    float* o = out + (size_t)t * (7 * B) + b;
    __builtin_nontemporal_store(sfW, o + 0 * B);
    __builtin_nontemporal_store(sfN, o + 1 * B);
    __builtin_nontemporal_store(sfR, o + 2 * B);
    __builtin_nontemporal_store(scW, o + 3 * B);
    __builtin_nontemporal_store(scN, o + 4 * B);
    __builtin_nontemporal_store(scR, o + 5 * B);
    __builtin_nontemporal_store(sh,  o + 6 * B);

    nu = nu_next;
  }
}

extern "C" void kernel_launch(void* const* d_in, const int* in_sizes, int n_in,
                              void* d_out, int out_size, void* d_ws, size_t ws_size,
                              hipStream_t stream) {
  (void)in_sizes; (void)n_in; (void)out_size; (void)d_ws; (void)ws_size;

  // setup_inputs() order: noise_input, g_RRe, g_RWe, g_WNi, g_WRi, g_NRi, g_NWi
  const float* noise = (const float*)d_in[0];
  const float* gRRe  = (const float*)d_in[1];
  const float* gRWe  = (const float*)d_in[2];
  const float* gWNi  = (const float*)d_in[3];
  const float* gWRi  = (const float*)d_in[4];
  const float* gNRi  = (const float*)d_in[5];
  const float* gNWi  = (const float*)d_in[6];
  float* out = (float*)d_out;

  // 512 lanes = 16 wave32s. One wave per workgroup -> 16 workgroups spread
  // across 16 WGPs, each wave alone on a SIMD32 (pure latency race).
  dim3 block(32);
  dim3 grid(srm::B / 32);
  SleepRegulationModel_88321707475483_kernel<<<grid, block, 0, stream>>>(
      noise, gRRe, gRWe, gWNi, gWRi, gNRi, gNWi, out);
}